// RoPEWithBetweenness_34333968564857
// MI455X (gfx1250) — compile-verified
//
#include <hip/hip_runtime.h>
#include <hip/hip_bf16.h>
#include <math.h>

typedef float v2f __attribute__((ext_vector_type(2)));
typedef float v8f __attribute__((ext_vector_type(8)));
typedef unsigned int u32x4 __attribute__((ext_vector_type(4)));
typedef int i32x4 __attribute__((ext_vector_type(4)));
typedef int i32x8 __attribute__((ext_vector_type(8)));

#define S_LEN 2048
#define H_NUM 16
#define D_DIM 64
#define TS 64            // bet values per block
#define RROWS 80         // TS + 16 halo rows (8 each side), multiple of 16
#define PITCH 66         // LDS row pitch (64 data + 2 pad DWORDs, via TDM pad)
#define CPITCH 65        // content pitch (odd -> conflict-free column walks)

// ---------------------------------------------------------------------------
// TDM: issue a 2D tensor_load_to_lds.
//   tile: tile_d1 rows x 64 fp32 elements, row stride `stride0` elements,
//   rows beyond tensor_d1 read as zero (OOB).  LDS padding: +2 DWORDs per
//   64-DWORD row => effective LDS pitch of 66 floats.
// ---------------------------------------------------------------------------
__device__ __forceinline__ void tdm_load_2d(unsigned lds_off, const float* gptr,
                                            unsigned tensor_d1, unsigned tile_d1,
                                            unsigned stride0) {
  const unsigned long long ga = (unsigned long long)(uintptr_t)gptr;
  u32x4 g0;
  g0[0] = 1u;                                   // count=1 (valid), user mode
  g0[1] = lds_off;                              // LDS byte address
  g0[2] = (unsigned)ga;                         // global_addr[31:0]
  g0[3] = (unsigned)((ga >> 32) & 0x1FFFFFFu)   // global_addr[56:32]
          | (2u << 30);                         // type=2 ("image")
  i32x8 g1;
  g1[0] = (int)((2u << 16)      // data_size = 4 bytes
                | (1u << 20)    // pad_enable
                | (5u << 22)    // pad_interval: every 64 DWORDs
                | (1u << 25));  // pad_amount: 2 DWORDs
  g1[1] = (int)(64u << 16);                           // tensor_dim0 = 64
  g1[2] = (int)(tensor_d1 << 16);                     // tensor_dim1[15:0]
  g1[3] = (int)((tensor_d1 >> 16) | (64u << 16));     // dim1 hi | tile_dim0 = 64
  g1[4] = (int)(tile_d1 & 0xFFFFu);                   // tile_dim1 (tile_dim2 = 0)
  g1[5] = (int)stride0;                               // tensor_dim0_stride
  g1[6] = 0;
  g1[7] = 0;
  const i32x4 z4 = {0, 0, 0, 0};
#if __has_include(<hip/amd_detail/amd_gfx1250_TDM.h>)
  const i32x8 z8 = {0, 0, 0, 0, 0, 0, 0, 0};
  __builtin_amdgcn_tensor_load_to_lds(g0, g1, z4, z4, z8, 0);
#else
  __builtin_amdgcn_tensor_load_to_lds(g0, g1, z4, z4, 0);
#endif
}

// ---------------------------------------------------------------------------
// Kernel 1: TDM-staged content-GEMM (fp32 WMMA) + distances + betweenness.
// One block = one (b,h) head and a 64-position chunk of the sequence.
// ---------------------------------------------------------------------------
__global__ __launch_bounds__(256) void bet_kernel(const float* __restrict__ x,
                                                  const float* __restrict__ W,
                                                  const float* __restrict__ bias,
                                                  float* __restrict__ betbuf) {
  const int chunk = blockIdx.x & 31;          // S/TS = 32 chunks
  const int bh    = blockIdx.x >> 5;          // 0..127
  const int b     = bh >> 4;
  const int h     = bh & 15;
  const int s0    = chunk * TS;
  const int r0    = s0 - 8;                   // first halo row (may be <0)
  const int tid   = threadIdx.x;
  const int wv    = tid >> 5;

  __shared__ float Wsh[64 * PITCH];           // Wsh[n*PITCH+k] = W[n][k]
  __shared__ float Xsh[RROWS * PITCH];
  __shared__ float bsh[64];
  __shared__ float Csh[RROWS * CPITCH];
  __shared__ float d1sh[TS + 1];              // d1[s0-1 .. s0+TS-1]
  __shared__ float d2sh[TS];                  // d2[s0-1 .. s0+TS-2]

  // --- Async staging via Tensor Data Mover (wave 0 issues; EXEC ignored) ---
  if (wv == 0) {
    // W: 64x64, contiguous rows (stride 64), padded to pitch 66 in LDS.
    tdm_load_2d((unsigned)(uintptr_t)&Wsh[0], W, 64u, 64u, 64u);
    // X tile: rows gr..gr+cnt-1 of this head (row stride H*D), landing at
    // local row (gr - r0).  Rows past S-1 are TDM-OOB -> zeros (unused).
    const int gr    = r0 < 0 ? 0 : r0;
    const int start = gr - r0;                // 0, or 8 for chunk 0
    const unsigned lds_off =
        (unsigned)(uintptr_t)&Xsh[0] + (unsigned)(start * PITCH * 4);
    const float* gsrc = x + (((size_t)b * S_LEN + gr) * H_NUM + h) * D_DIM;
    tdm_load_2d(lds_off, gsrc, (unsigned)(S_LEN - gr),
                (unsigned)(RROWS - start), (unsigned)(H_NUM * D_DIM));
  }
  if (tid < 64) bsh[tid] = bias[tid];
  if (wv == 0) __builtin_amdgcn_s_wait_tensorcnt(0);
  __syncthreads();

  // --- GEMM: content[RROWS x 64] = X * W^T + bias via v_wmma_f32_16x16x4 ---
  const int lane = tid & 31;
  const int m    = lane & 15;
  const int hi16 = lane >> 4;                 // 0 for lanes 0-15, 1 for 16-31
  for (int t = wv; t < (RROWS / 16) * 4; t += 8) {
    const int mt = t >> 2;
    const int nt = t & 3;
    const int n  = nt * 16 + m;
    v8f acc;
    const float bv = bsh[n];
#pragma unroll
    for (int r = 0; r < 8; ++r) acc[r] = bv;
#pragma unroll
    for (int ki = 0; ki < 16; ++ki) {
      const int klo  = ki * 4 + hi16 * 2;
      const int arow = mt * 16 + m;
      v2f a;
      a[0] = Xsh[arow * PITCH + klo];
      a[1] = Xsh[arow * PITCH + klo + 1];
      v2f bb;
      bb[0] = Wsh[n * PITCH + klo];           // B[k][n] = W[n][k]
      bb[1] = Wsh[n * PITCH + klo + 1];
      acc = __builtin_amdgcn_wmma_f32_16x16x4_f32(
          /*neg_a=*/false, a, /*neg_b=*/false, bb,
          /*c_mod=*/(short)0, acc, /*reuse_a=*/false, /*reuse_b=*/false);
    }
#pragma unroll
    for (int r = 0; r < 8; ++r) {
      const int row = mt * 16 + r + hi16 * 8;
      Csh[row * CPITCH + n] = acc[r];
    }
  }
  __syncthreads();

  // --- Neighbor distances: d1[i]=||C[g+1]-C[g]||, d2[i]=||C[g+2]-C[g]||,
  //     g = s0-1+i (local row 7+i). ---
  if (tid < TS + 1) {
    const float* p0 = &Csh[(7 + tid) * CPITCH];
    const float* p1 = &Csh[(8 + tid) * CPITCH];
    float acc = 0.0f;
    for (int k = 0; k < 64; ++k) { const float d = p1[k] - p0[k]; acc += d * d; }
    d1sh[tid] = sqrtf(acc);
  } else if (tid < 2 * TS + 1) {
    const int i2 = tid - (TS + 1);
    const float* p0 = &Csh[(7 + i2) * CPITCH];
    const float* p1 = &Csh[(9 + i2) * CPITCH];
    float acc = 0.0f;
    for (int k = 0; k < 64; ++k) { const float d = p1[k] - p0[k]; acc += d * d; }
    d2sh[i2] = sqrtf(acc);
  }
  __syncthreads();

  // --- Betweenness: relu(1 - (path-direct)/max(direct,1e-6)) / (S-2) ---
  if (tid < TS) {
    const int s = s0 + tid;
    float bet = 0.0f;
    if (s > 0 && s < S_LEN - 1) {
      const float path   = d1sh[tid] + d1sh[tid + 1];   // d1[s-1] + d1[s]
      const float direct = d2sh[tid];                   // d2[s-1]
      const float rel    = (path - direct) / fmaxf(direct, 1e-6f);
      bet = fmaxf(0.0f, 1.0f - rel) * (1.0f / (float)(S_LEN - 2));
    }
    betbuf[(size_t)bh * S_LEN + s] = bet;
  }
}

// ---------------------------------------------------------------------------
// Kernel 2: interpolated-RoPE apply.  One thread per (j, j+32) pair; fully
// coalesced, bandwidth-bound (sets the ~7us HBM floor at 23.3 TB/s).
// ---------------------------------------------------------------------------
__global__ __launch_bounds__(256) void rope_kernel(const float* __restrict__ x,
                                                   const float* __restrict__ betbuf,
                                                   const float* __restrict__ gatep,
                                                   float* __restrict__ out) {
  const size_t p = (size_t)blockIdx.x * blockDim.x + threadIdx.x;   // B*S*H*32
  const int j      = (int)(p & 31);
  const size_t row = p >> 5;                 // ((b*S + s)*H + h)
  const int h = (int)(row & 15);
  const int s = (int)((row >> 4) & 2047);
  const int b = (int)(row >> 15);

  const float bet  = betbuf[(((size_t)b * H_NUM + h) * S_LEN) + s];
  const float gate = *gatep;

  float pos = (float)s + gate * (bet - 0.5f) * 0.1f;   // ADJ_SCALE = 0.1
  pos = fminf(fmaxf(pos, 0.0f), (float)(S_LEN - 1));   // MAX_SEQ_LEN-1 == S-1
  const float lo   = floorf(pos);
  const float hicl = ceilf(pos);
  const float frac = pos - lo;
  const float ipos = lo + frac * (hicl - lo);          // == pos (matches ref)

  // inv_freq[j] = 10000^(-2j/64) = exp(-ln(10000)/32 * j)
  const float invf = __expf(-0.28782313662425574f * (float)j);
  const float ang  = ipos * invf;
  const float c  = cosf(ang);
  const float sn = sinf(ang);

  const float x1 = x[row * 64 + j];
  const float x2 = x[row * 64 + j + 32];
  out[row * 64 + j]      = x1 * c + x2 * sn;
  out[row * 64 + j + 32] = x2 * c + x1 * sn;
}

extern "C" void kernel_launch(void* const* d_in, const int* in_sizes, int n_in,
                              void* d_out, int out_size, void* d_ws, size_t ws_size,
                              hipStream_t stream) {
  const float* x    = (const float*)d_in[0];
  const float* W    = (const float*)d_in[1];
  const float* bias = (const float*)d_in[2];
  const float* gate = (const float*)d_in[3];
  float* out    = (float*)d_out;
  float* betbuf = (float*)d_ws;              // needs B*H*S*4 = 1 MB

  // Kernel 1: 128 heads * 32 chunks = 4096 blocks.
  bet_kernel<<<128 * (S_LEN / TS), 256, 0, stream>>>(x, W, bias, betbuf);

  // Kernel 2: B*S*H*32 threads = 8,388,608 -> 32768 blocks.
  rope_kernel<<<(8 * S_LEN * H_NUM * 32) / 256, 256, 0, stream>>>(x, betbuf, gate, out);
}